// BioSphericalCKN1D_48421461295423
// MI455X (gfx1250) — compile-verified
//
#include <hip/hip_runtime.h>
#include <hip/hip_bf16.h>

typedef __attribute__((ext_vector_type(2))) float v2f;
typedef __attribute__((ext_vector_type(4))) float v4f;
typedef __attribute__((ext_vector_type(8))) float v8f;

#define K_TAPS 15
#define CIN    128
#define F_OUT  256
#define L_IN   4096
#define L_OUT  (L_IN - K_TAPS + 1)     // 4082
#define BATCH  32
#define TILE_M 64                       // output positions per block
#define ROWS   (TILE_M + K_TAPS - 1)    // 78 raw x rows per block
#define XPAD   132                      // x LDS row stride (floats): 4-bank/row skew -> conflict-free A gathers
#define KC     32                       // K-chunk staged per refill; divides a tap (128) exactly
#define KPAD   36                       // transposed-B row stride: lanes 0-15 hit banks 0 mod 4,
                                        // lanes 16-31 (+2 dwords) hit banks 2 mod 4 -> conflict-free b64 reads
#define NCHUNK ((K_TAPS * CIN) / KC)    // 60
#define EPS_F  1e-7f

__global__ __launch_bounds__(256)
void ckn1d_wmma_kernel(const float* __restrict__ x,
                       const float* __restrict__ w,      // (K_TAPS*CIN, F_OUT) row-major
                       const float* __restrict__ scale,
                       const float* __restrict__ bias,
                       float* __restrict__ out) {
    __shared__ float lds_x[ROWS * XPAD];      // 41184 B: x slab (zero-padded past L_IN)
    __shared__ float lds_bt[F_OUT * KPAD];    // 36864 B: weight chunk, TRANSPOSED [f][kc]
    __shared__ float lds_rowsq[ROWS];
    __shared__ float lds_invn[TILE_M];
    __shared__ float lds_bias[F_OUT];

    const int t    = threadIdx.x;
    const int lane = t & 31;
    const int wave = t >> 5;
    const int l0   = blockIdx.x * TILE_M;
    const int b    = blockIdx.y;

    // ---- 1) Stage x[b, l0 .. l0+77, :] into LDS (coalesced float4, zero-fill OOB) ----
    const float* xb = x + (size_t)b * L_IN * CIN;
    #pragma unroll
    for (int i = 0; i < 10; ++i) {
        int q = t + 256 * i;                  // float4 index; 78*32 = 2496 total
        if (q < ROWS * (CIN / 4)) {
            int row = q >> 5;
            int f4  = q & 31;
            int g   = l0 + row;
            v4f val = {0.f, 0.f, 0.f, 0.f};
            if (g < L_IN) val = *(const v4f*)(xb + (size_t)g * CIN + f4 * 4);
            *(v4f*)(&lds_x[row * XPAD + f4 * 4]) = val;
        }
    }
    if (t < F_OUT) lds_bias[t] = bias[t];
    __syncthreads();

    // ---- 2) Sliding sum-of-squares -> fused epilogue scale (invn) ----
    if (t < ROWS) {
        const v4f* r = (const v4f*)&lds_x[t * XPAD];   // 528 B row stride, 16B aligned
        float s = 0.f;
        #pragma unroll
        for (int c = 0; c < CIN / 4; ++c) {
            v4f v = r[c];
            s += v.x * v.x + v.y * v.y + v.z * v.z + v.w * v.w;
        }
        lds_rowsq[t] = s;
    }
    __syncthreads();
    if (t < TILE_M) {
        float win = 0.f;
        #pragma unroll
        for (int k = 0; k < K_TAPS; ++k) win += lds_rowsq[t + k];
        lds_invn[t] = scale[0] / __builtin_sqrtf(win + EPS_F);
    }
    // lds_invn visibility guaranteed by the first barrier inside the chunk loop

    // ---- 3) Implicit-GEMM main loop: v_wmma_f32_16x16x4_f32, register-prefetched B chunks ----
    const int mt   = wave & 3;                // M sub-tile 0..3 (16 positions each)
    const int f0   = (wave >> 2) * 128;       // N half: filters [f0, f0+128)
    const int half = lane >> 4;               // 0: lanes 0-15 (K=0,1) | 1: lanes 16-31 (K=2,3)
    const int lm   = lane & 15;

    v8f acc[8] = {};                          // 16x128 f32 output tile per wave

    // Prefetch chunk 0 into registers: thread t owns weight column f = t,
    // K pair (2i, 2i+1). Global reads are lane-consecutive in f (coalesced dwords).
    float pa[16], pb[16];
    {
        const float* src = w;
        #pragma unroll
        for (int i = 0; i < 16; ++i) {
            pa[i] = src[(2 * i)     * F_OUT + t];
            pb[i] = src[(2 * i + 1) * F_OUT + t];
        }
    }

    for (int ch = 0; ch < NCHUNK; ++ch) {
        __syncthreads();                      // previous chunk's consumers done with lds_bt
        {   // transpose-store current chunk: one b64 per K pair at [f][kc]
            float* dst = &lds_bt[t * KPAD];
            #pragma unroll
            for (int i = 0; i < 16; ++i) {
                v2f p;
                p.x = pa[i];
                p.y = pb[i];
                *(v2f*)(dst + 2 * i) = p;
            }
        }
        if (ch + 1 < NCHUNK) {                // issue next chunk's global loads; latency
            const float* src = w + (size_t)(ch + 1) * KC * F_OUT;   // hides under the WMMAs below
            #pragma unroll
            for (int i = 0; i < 16; ++i) {
                pa[i] = src[(2 * i)     * F_OUT + t];
                pb[i] = src[(2 * i + 1) * F_OUT + t];
            }
        }
        __syncthreads();                      // lds_bt ready

        const int tap = ch >> 2;              // conv tap this chunk belongs to
        const int c0  = (ch & 3) * KC;        // channel base inside the tap
        const float* arow = &lds_x[(mt * 16 + lm + tap) * XPAD + c0 + 2 * half];
        const float* bcol = &lds_bt[(f0 + lm) * KPAD + 2 * half];

        #pragma unroll
        for (int kc = 0; kc < KC; kc += 4) {
            v2f a = *(const v2f*)(arow + kc);                       // conflict-free b64
            #pragma unroll
            for (int nt = 0; nt < 8; ++nt) {
                v2f bf = *(const v2f*)(bcol + nt * 16 * KPAD + kc); // single b64, no repack
                acc[nt] = __builtin_amdgcn_wmma_f32_16x16x4_f32(
                    false, a, false, bf, (short)0, acc[nt], false, false);
            }
        }
    }

    // ---- 4) Epilogue: out = dot * invn + bias, masked for the L tail ----
    float* ob = out + ((size_t)b * L_OUT + l0) * F_OUT;
    #pragma unroll
    for (int v = 0; v < 8; ++v) {
        int mrow = mt * 16 + v + 8 * half;    // C layout: VGPR v -> M = v (lanes 0-15) / v+8 (16-31)
        if (l0 + mrow < L_OUT) {
            float inv = lds_invn[mrow];
            #pragma unroll
            for (int nt = 0; nt < 8; ++nt) {
                int f = f0 + nt * 16 + lm;
                ob[(size_t)mrow * F_OUT + f] = acc[nt][v] * inv + lds_bias[f];
            }
        }
    }
}

extern "C" void kernel_launch(void* const* d_in, const int* in_sizes, int n_in,
                              void* d_out, int out_size, void* d_ws, size_t ws_size,
                              hipStream_t stream) {
    const float* x     = (const float*)d_in[0];   // (32, 4096, 128)
    const float* w     = (const float*)d_in[1];   // (15, 128, 256) == (1920, 256)
    const float* scale = (const float*)d_in[2];   // (1,)
    const float* bias  = (const float*)d_in[3];   // (256,)
    float* out = (float*)d_out;                   // (32, 4082, 256)

    dim3 grid((L_OUT + TILE_M - 1) / TILE_M, BATCH);  // (64, 32)
    ckn1d_wmma_kernel<<<grid, 256, 0, stream>>>(x, w, scale, bias, out);
}